// SoftDTW_86560770884099
// MI455X (gfx1250) — compile-verified
//
#include <hip/hip_runtime.h>
#include <hip/hip_bf16.h>
#include <math.h>

typedef float v2f __attribute__((ext_vector_type(2)));
typedef float v8f __attribute__((ext_vector_type(8)));

#define BATCH 32
#define NN 1024
#define MM 1024
#define KK 64
#define BIGV 10000000000.0f

// ---------------------------------------------------------------------------
// Kernel A: batched S[b,n,m] = sum_k X[b,n,k] * Y[b,m,k]  via V_WMMA_F32_16X16X4_F32
// Block = 256 threads (8 waves). Each block covers a 16 x 128 output strip of
// one batch; each wave owns one 16x16 tile and iterates K=64 in 16 WMMA steps.
// All 8 waves share the same 16-row A strip -> X loads hit L1.
// ---------------------------------------------------------------------------
__global__ __launch_bounds__(256) void softdtw_gemm_wmma(const float* __restrict__ X,
                                                         const float* __restrict__ Y,
                                                         float* __restrict__ S) {
    const int wave = threadIdx.x >> 5;
    const int lane = threadIdx.x & 31;
    const int half = lane >> 4;       // 0: lanes 0-15, 1: lanes 16-31
    const int lm   = lane & 15;

    const int tjBlocks = MM / 128;                     // 8
    const int tiBlocks = NN / 16;                      // 64
    int tj_blk = blockIdx.x % tjBlocks;
    int ti     = (blockIdx.x / tjBlocks) % tiBlocks;
    int b      = blockIdx.x / (tjBlocks * tiBlocks);

    const int i0 = ti * 16;
    const int j0 = tj_blk * 128 + wave * 16;

    // A: 16x4 f32 tile. Lane (half,lm): VGPR pair = X[i0+lm, k0+2*half + {0,1}]
    // B: 4x16 f32 tile. Lane (half,lm): VGPR pair = Y[j0+lm, k0+2*half + {0,1}]
    const float* Xrow = X + ((size_t)b * NN + i0 + lm) * KK;
    const float* Yrow = Y + ((size_t)b * MM + j0 + lm) * KK;

    v8f c = {};
#pragma unroll
    for (int k0 = 0; k0 < KK; k0 += 4) {
        v2f a  = *(const v2f*)(Xrow + k0 + 2 * half);
        v2f bb = *(const v2f*)(Yrow + k0 + 2 * half);
        // 8 args: (neg_a, A, neg_b, B, c_mod, C, reuse_a, reuse_b)
        c = __builtin_amdgcn_wmma_f32_16x16x4_f32(false, a, false, bb,
                                                  (short)0, c, false, false);
    }

    // C/D layout: VGPR v -> row (v + 8*half), col lm. Coalesced 64B row stores.
    float* Sb = S + ((size_t)b * NN + i0) * MM + j0;
#pragma unroll
    for (int v = 0; v < 8; ++v) {
        int row = v + 8 * half;
        Sb[(size_t)row * MM + lm] = c[v];
    }
}

// ---------------------------------------------------------------------------
// Kernel B: per-batch anti-diagonal soft-DTW DP, fully inside one workgroup.
// 1024 threads; thread t owns DP index a = t+1. Three rolling 1025-entry rows
// in LDS; norms of X/Y rows staged in LDS once.  2047 barrier-synced steps.
// ---------------------------------------------------------------------------
__global__ __launch_bounds__(1024) void softdtw_dp(const float* __restrict__ X,
                                                   const float* __restrict__ Y,
                                                   const float* __restrict__ S,
                                                   float* __restrict__ out) {
    const int b = blockIdx.x;
    const int t = threadIdx.x;          // 0..1023

    __shared__ float xx[NN];
    __shared__ float yy[MM];
    __shared__ float rbuf[3][NN + 2];

    // Row norms: thread t does ||X[b,t,:]||^2 and ||Y[b,t,:]||^2 (64 MACs each)
    {
        const float* xr = X + ((size_t)b * NN + t) * KK;
        const float* yr = Y + ((size_t)b * MM + t) * KK;
        float sx = 0.f, sy = 0.f;
#pragma unroll 8
        for (int k = 0; k < KK; ++k) {
            sx = fmaf(xr[k], xr[k], sx);
            sy = fmaf(yr[k], yr[k], sy);
        }
        xx[t] = sx;
        yy[t] = sy;
    }

    // r0: [0]=0, rest BIG.  r1: all BIG.
    rbuf[0][t + 1] = BIGV;
    rbuf[1][t + 1] = BIGV;
    if (t == 0) {
        rbuf[0][0] = 0.0f;
        rbuf[1][0] = BIGV;
        rbuf[2][0] = BIGV;
    }
    __syncthreads();

    const float* Sb = S + (size_t)b * NN * MM;

    int idx = 2;                        // which buffer holds the new row
    const int a = t + 1;                // DP index owned by this thread
    for (int k = 2; k <= NN + MM; ++k) {
        float*       r   = rbuf[idx];
        const float* rp1 = rbuf[(idx + 2) % 3];   // one step back
        const float* rp2 = rbuf[(idx + 1) % 3];   // two steps back

        const int i = a - 1;            // = t
        const int j = k - a - 1;        // column into S

        float val = BIGV;
        if (j >= 0 && j < MM) {
            float d  = xx[i] + yy[j] - 2.0f * Sb[(size_t)i * MM + j];
            float x1 = rp2[a - 1];      // diag
            float x2 = rp1[a - 1];      // up
            float x3 = rp1[a];          // left
            float m  = fminf(x1, fminf(x2, x3));
            float s  = __expf(m - x1) + __expf(m - x2) + __expf(m - x3);
            val = d + (m - __logf(s));  // gamma = 1
        }
        r[a] = val;
        if (t == 0) r[0] = BIGV;
        __syncthreads();

        if (k == NN + MM && t == 0) out[b] = r[NN];

        idx = (idx + 1) % 3;
    }
}

// ---------------------------------------------------------------------------
extern "C" void kernel_launch(void* const* d_in, const int* in_sizes, int n_in,
                              void* d_out, int out_size, void* d_ws, size_t ws_size,
                              hipStream_t stream) {
    const float* X = (const float*)d_in[0];   // [32,1024,64] f32
    const float* Y = (const float*)d_in[1];   // [32,1024,64] f32
    float* out = (float*)d_out;               // [32] f32
    float* S   = (float*)d_ws;                // needs 32*1024*1024*4 = 128 MB

    // Phase 1: WMMA batched X.Y^T
    dim3 grid1(BATCH * (NN / 16) * (MM / 128));
    softdtw_gemm_wmma<<<grid1, 256, 0, stream>>>(X, Y, S);

    // Phase 2: per-batch wavefront DP (one workgroup per batch)
    softdtw_dp<<<BATCH, 1024, 0, stream>>>(X, Y, S, out);
}